// FreqLAWDS_80771154968836
// MI455X (gfx1250) — compile-verified
//
#include <hip/hip_runtime.h>

typedef float v2f __attribute__((ext_vector_type(2)));
typedef float v8f __attribute__((ext_vector_type(8)));

// 16-byte int vector type matching the async-LDS builtin's expected pointee
typedef int fls_v4i __attribute__((vector_size(4 * sizeof(int))));
typedef __attribute__((address_space(1))) fls_v4i* fls_gv4p;  // global
typedef __attribute__((address_space(3))) fls_v4i* fls_lv4p;  // LDS

#define FLS_HAS_ASYNC_LDS (__has_builtin(__builtin_amdgcn_global_load_async_to_lds_b128) && \
                           __has_builtin(__builtin_amdgcn_s_wait_asynccnt))

__device__ __forceinline__ float silu_f(float y) {
    return y * (1.0f / (1.0f + __expf(-y)));
}

// ---------------------------------------------------------------------------
// Cooperative stage of a contiguous, 16B-aligned float region into LDS.
// Uses gfx1250 async global->LDS copies (ASYNCcnt) when the builtin exists;
// falls back to plain load+ds_store otherwise. Assumes blockDim.x == 256.
// ---------------------------------------------------------------------------
__device__ __forceinline__ void fls_stage_lds(float* smem, const float* __restrict__ gsrc,
                                              int nfloats) {
    const int tid = threadIdx.x;
    const int nvec = nfloats >> 2;
    const float4* g4 = (const float4*)gsrc;
    float4* s4 = (float4*)smem;
#if FLS_HAS_ASYNC_LDS
    for (int i = tid; i < nvec; i += 256) {
        __builtin_amdgcn_global_load_async_to_lds_b128(
            (fls_gv4p)(g4 + i), (fls_lv4p)(s4 + i), 0, 0);
    }
    __builtin_amdgcn_s_wait_asynccnt(0);
    __syncthreads();
#else
    for (int i = tid; i < nvec; i += 256) s4[i] = g4[i];
    __syncthreads();
#endif
}

// ---------------------------------------------------------------------------
// 3x3 avg pool, stride 1, pad 1, count_include_pad (always /9)
// ---------------------------------------------------------------------------
__global__ void fls_pool3_kernel(const float* __restrict__ x, float* __restrict__ out,
                                 long total) {
    long idx = (long)blockIdx.x * blockDim.x + threadIdx.x;
    if (idx >= total) return;
    int xo = (int)(idx & 255);
    int yo = (int)((idx >> 8) & 255);
    long bc = idx >> 16;
    const float* p = x + (bc << 16);
    float s = 0.0f;
    #pragma unroll
    for (int dy = -1; dy <= 1; ++dy) {
        int yy = yo + dy;
        if (yy < 0 || yy >= 256) continue;
        #pragma unroll
        for (int dx = -1; dx <= 1; ++dx) {
            int xx = xo + dx;
            if (xx < 0 || xx >= 256) continue;
            s += p[yy * 256 + xx];
        }
    }
    out[idx] = s * (1.0f / 9.0f);
}

// ---------------------------------------------------------------------------
// Haar subpixel decomposition at 128x128:
//   llhf    : (B,256,hw)  ch [0,128)=LL, ch [128,256)=|H1|+|H2|+|H3|
//   freq_ctx: (B,384,hw)  ch [256,384) <- coarse (avgpool2x2)
// subpixel order (a,b,c,d) = (0,0),(1,0),(0,1),(1,1)
// ---------------------------------------------------------------------------
__global__ void fls_haar_kernel(const float* __restrict__ x,
                                float* __restrict__ llhf,
                                float* __restrict__ freq_ctx, long total) {
    long idx = (long)blockIdx.x * blockDim.x + threadIdx.x;
    if (idx >= total) return;
    const int hw = 128 * 128;
    int xo = (int)(idx & 127);
    int yo = (int)((idx >> 7) & 127);
    int c  = (int)((idx >> 14) & 127);
    int b  = (int)(idx >> 21);
    const float* xb = x + (((long)(b * 128 + c) * 256 + 2 * yo) * 256 + 2 * xo);
    float a  = xb[0];
    float bb = xb[256];
    float cc = xb[1];
    float dd = xb[257];
    float ll = 0.5f * (a + bb + cc + dd);
    float h1 = 0.5f * (-a + bb - cc + dd);
    float h2 = 0.5f * (-a - bb + cc + dd);
    float h3 = 0.5f * (a - bb - cc + dd);
    float hf = fabsf(h1) + fabsf(h2) + fabsf(h3);
    float coarse = 0.25f * (a + bb + cc + dd);
    long pix = (long)yo * 128 + xo;
    llhf[((long)b * 256 + c) * hw + pix]       = ll;
    llhf[((long)b * 256 + 128 + c) * hw + pix] = hf;
    freq_ctx[((long)b * 384 + 256 + c) * hw + pix] = coarse;
}

// ---------------------------------------------------------------------------
// Dense GEMM (1x1 conv): D[M,N] = W[M,K] @ Xe[K,N], per batch (blockIdx.z).
// Xe = X1 (+ alpha*X2 when HAS_X2). Epilogue: y = d*scale[m]+bias[m]; SiLU opt.
// Weight tile (64 rows x K) staged in LDS (async). Block = 8 wave32s; each
// wave = 32M x 32N macro-tile: 2 A frags (one ds_load_2addr) x 2 B frags ->
// 4 WMMA per k-step. B addressed as uniform base + loop-invariant 32-bit lane
// offset so codegen uses SADDR+voffset global loads (1 s_add per k-step).
// Grid: x = N/128, y = M/64, z = batch.
// ---------------------------------------------------------------------------
template<int K, bool HAS_X2>
__global__ void __launch_bounds__(256)
fls_wmma_gemm_cbs(const float* __restrict__ Wt, const float* __restrict__ X1,
                  const float* __restrict__ X2, const float* __restrict__ alphaPtr,
                  const float* __restrict__ scale, const float* __restrict__ bias,
                  float* __restrict__ Y, int M, int N, int do_silu) {
    __shared__ float sA[64 * K];
    const int lane = threadIdx.x & 31;
    const int wv   = threadIdx.x >> 5;
    const int lr   = lane & 15;
    const int hi   = lane >> 4;   // 0: M=r | K0,K1 ; 1: M=r+8 | K2,K3
    const int koff = hi * 2;
    const int mBlock = blockIdx.y * 64;
    const int mBase  = (wv >> 2) * 32;
    const int n0     = blockIdx.x * 128 + (wv & 3) * 32;
    const long xoff  = (long)blockIdx.z * K * N;
    const long yoff  = (long)blockIdx.z * M * N;
    const float alpha = HAS_X2 ? alphaPtr[0] : 0.0f;

    fls_stage_lds(sA, Wt + (long)mBlock * K, 64 * K);

    v8f acc[2][2] = {};
    const int col  = n0 + lr;
    const int lofs = koff * N + col;          // per-lane, loop-invariant (32-bit)
    const float* sA0 = &sA[(mBase + lr) * K + koff];
    const float* sA1 = sA0 + 16 * K;
    const float* Xk  = X1 + xoff;             // wave-uniform k-row base
    const float* X2k = HAS_X2 ? (X2 + xoff) : nullptr;
    #pragma unroll 2
    for (int k0 = 0; k0 < K; k0 += 4) {
        v2f a0 = *(const v2f*)(sA0 + k0);     // one ds_load_2addr_b64 pair
        v2f a1 = *(const v2f*)(sA1 + k0);
        __builtin_prefetch(Xk + 8 * (size_t)N + lofs, 0, 1);  // 2 k-steps ahead
        float p00 = Xk[lofs];
        float p01 = Xk[lofs + N];
        float p10 = Xk[lofs + 16];
        float p11 = Xk[lofs + N + 16];
        if (HAS_X2) {
            p00 += alpha * X2k[lofs];
            p01 += alpha * X2k[lofs + N];
            p10 += alpha * X2k[lofs + 16];
            p11 += alpha * X2k[lofs + N + 16];
        }
        v2f b0, b1;
        b0.x = p00; b0.y = p01;
        b1.x = p10; b1.y = p11;
        acc[0][0] = __builtin_amdgcn_wmma_f32_16x16x4_f32(false, a0, false, b0, (short)0, acc[0][0], false, false);
        acc[0][1] = __builtin_amdgcn_wmma_f32_16x16x4_f32(false, a0, false, b1, (short)0, acc[0][1], false, false);
        acc[1][0] = __builtin_amdgcn_wmma_f32_16x16x4_f32(false, a1, false, b0, (short)0, acc[1][0], false, false);
        acc[1][1] = __builtin_amdgcn_wmma_f32_16x16x4_f32(false, a1, false, b1, (short)0, acc[1][1], false, false);
        Xk += 4 * (size_t)N;
        if (HAS_X2) X2k += 4 * (size_t)N;
    }
    #pragma unroll
    for (int mm = 0; mm < 2; ++mm) {
        #pragma unroll
        for (int r = 0; r < 8; ++r) {
            int row = mBlock + mBase + mm * 16 + r + hi * 8;
            float s  = (scale != nullptr) ? scale[row] : 1.0f;
            float bb = bias[row];
            float v0 = acc[mm][0][r] * s + bb;
            float v1 = acc[mm][1][r] * s + bb;
            if (do_silu) { v0 = silu_f(v0); v1 = silu_f(v1); }
            long yb = yoff + (long)row * N + col;
            Y[yb]      = v0;
            Y[yb + 16] = v1;
        }
    }
}

// ---------------------------------------------------------------------------
// Grouped conv as per-group GEMM with on-the-fly im2col B gather.
// Compile-time (CoutG, KS, S, P): divisions become mul-shift, pad check
// vanishes for 1x1. Weight tile (CoutG x K) staged in LDS (async).
// B gathers addressed as wave-uniform base + 32-bit per-lane offset.
// Wave = (MW m-tiles) x (2 n-tiles); MW=2 when CoutG>=32.
// Grid: x = N/(NWAVES*32), y = group, z = batch.
// ---------------------------------------------------------------------------
template<int CoutG, int KS, int S, int P>
__global__ void __launch_bounds__(256)
fls_wmma_gconv_cbs(const float* __restrict__ Wt, const float* __restrict__ X,
                   int CinTot, int Hin, int Win,
                   const float* __restrict__ scale, const float* __restrict__ bias,
                   float* __restrict__ Y, int CoutTotDest, int ocOffset,
                   int Ho, int Wo, int do_silu) {
    constexpr int CinG = 16;
    constexpr int K    = CinG * KS * KS;
    constexpr int KS2  = KS * KS;
    constexpr int MT   = CoutG / 16;          // m-tiles total (1 or 4)
    constexpr int MW   = (MT >= 2) ? 2 : 1;   // m-tiles per wave
    constexpr int MWAVES = MT / MW;           // 2 or 1
    constexpr int NWAVES = 8 / MWAVES;        // 4 or 8
    constexpr bool CHK = (P > 0);
    __shared__ float sW[CoutG * K];

    const int lane = threadIdx.x & 31;
    const int wv   = threadIdx.x >> 5;
    const int lr   = lane & 15;
    const int hi   = lane >> 4;
    const int koff = hi * 2;
    const int mi = wv % MWAVES;
    const int ni = wv / MWAVES;
    const int g = blockIdx.y;
    const int b = blockIdx.z;
    const int mBase = mi * MW * 16;
    const int n0 = (blockIdx.x * NWAVES + ni) * 32;

    fls_stage_lds(sW, Wt + (long)g * CoutG * K, CoutG * K);

    const int HinWin = Hin * Win;
    const float* xbase = X + ((long)b * CinTot + g * CinG) * HinWin;  // uniform
    const int nA = n0 + lr;
    const int nB = nA + 16;
    const int oyA = nA / Wo, oxA = nA - oyA * Wo;
    const int oyB = nB / Wo, oxB = nB - oyB * Wo;
    const int iyA0 = oyA * S - P, ixA0 = oxA * S - P;
    const int iyB0 = oyB * S - P, ixB0 = oxB * S - P;

    v8f acc[MW][2] = {};
    const float* sW0 = &sW[(mBase + lr) * K + koff];
    #pragma unroll 4
    for (int k0 = 0; k0 < K; k0 += 4) {
        v2f a[MW];
        #pragma unroll
        for (int mm = 0; mm < MW; ++mm)
            a[mm] = *(const v2f*)(sW0 + mm * 16 * K + k0);   // ds_load_b64
        v2f bf[2];
        #pragma unroll
        for (int e = 0; e < 2; ++e) {
            int k   = k0 + koff + e;
            int ic  = k / KS2;          // constant divisor -> mul/shift
            int rem = k - ic * KS2;
            int ky  = rem / KS;
            int kx  = rem - ky * KS;
            const int cofs = ic * HinWin;
            {
                int iy = iyA0 + ky, ix = ixA0 + kx;
                float v = 0.0f;
                if (!CHK || ((unsigned)iy < (unsigned)Hin && (unsigned)ix < (unsigned)Win))
                    v = xbase[cofs + iy * Win + ix];
                if (e == 0) bf[0].x = v; else bf[0].y = v;
            }
            {
                int iy = iyB0 + ky, ix = ixB0 + kx;
                float v = 0.0f;
                if (!CHK || ((unsigned)iy < (unsigned)Hin && (unsigned)ix < (unsigned)Win))
                    v = xbase[cofs + iy * Win + ix];
                if (e == 0) bf[1].x = v; else bf[1].y = v;
            }
        }
        #pragma unroll
        for (int mm = 0; mm < MW; ++mm) {
            acc[mm][0] = __builtin_amdgcn_wmma_f32_16x16x4_f32(false, a[mm], false, bf[0], (short)0, acc[mm][0], false, false);
            acc[mm][1] = __builtin_amdgcn_wmma_f32_16x16x4_f32(false, a[mm], false, bf[1], (short)0, acc[mm][1], false, false);
        }
    }
    const long hwl = (long)Ho * Wo;
    const long ybatch = (long)b * CoutTotDest * hwl;
    #pragma unroll
    for (int mm = 0; mm < MW; ++mm) {
        #pragma unroll
        for (int r = 0; r < 8; ++r) {
            int oc = g * CoutG + mBase + mm * 16 + r + hi * 8;
            float s0 = (scale != nullptr) ? scale[oc] : 1.0f;
            float bb = bias[oc];
            float v0 = acc[mm][0][r] * s0 + bb;
            float v1 = acc[mm][1][r] * s0 + bb;
            if (do_silu) { v0 = silu_f(v0); v1 = silu_f(v1); }
            long yb = ybatch + (long)(ocOffset + oc) * hwl;
            Y[yb + nA] = v0;
            Y[yb + nB] = v1;
        }
    }
}

// ---------------------------------------------------------------------------
// Attention fuse: subpixel-gather local logits from la_out (256x256), compute
// fg2 grouped 1x1 (16 MACs) in VALU on fg1_out, softmax over 4 candidates,
// weighted sum of cand -> fused (B,128,128,128).
// ---------------------------------------------------------------------------
__global__ void fls_att_fuse_kernel(const float* __restrict__ la_out,
                                    const float* __restrict__ fg1,
                                    const float* __restrict__ cand,
                                    const float* __restrict__ fg2_w,
                                    const float* __restrict__ fg2_b,
                                    const float* __restrict__ fsPtr,
                                    float* __restrict__ fused, long total) {
    long idx = (long)blockIdx.x * blockDim.x + threadIdx.x;
    if (idx >= total) return;
    const int hw = 128 * 128;
    int xo = (int)(idx & 127);
    int yo = (int)((idx >> 7) & 127);
    int c  = (int)((idx >> 14) & 127);
    int b  = (int)(idx >> 21);
    float fs = fsPtr[0];
    const float* lb = la_out + (((long)(b * 128 + c) * 256 + 2 * yo) * 256 + 2 * xo);
    float lg[4];
    lg[0] = lb[0];
    lg[1] = lb[256];
    lg[2] = lb[1];
    lg[3] = lb[257];
    long pix = (long)yo * 128 + xo;
    const float* fg1b = fg1 + (long)b * 128 * hw + pix;
    float logit[4];
    float m = -1e30f;
    #pragma unroll
    for (int j = 0; j < 4; ++j) {
        int oc = j * 128 + c;
        int grp = oc >> 6;
        const float* wv = fg2_w + oc * 16;
        const float* xin = fg1b + (long)(grp * 16) * hw;
        float a = fg2_b[oc];
        #pragma unroll
        for (int k = 0; k < 16; ++k) a += wv[k] * xin[(long)k * hw];
        logit[j] = lg[j] + fs * a;
        m = fmaxf(m, logit[j]);
    }
    float sum = 0.0f;
    #pragma unroll
    for (int j = 0; j < 4; ++j) {
        logit[j] = __expf(logit[j] - m);
        sum += logit[j];
    }
    float inv = 1.0f / sum;
    const float* cb = cand + (long)b * 512 * hw + pix;
    float f = 0.0f;
    #pragma unroll
    for (int j = 0; j < 4; ++j)
        f += cb[(long)(j * 128 + c) * hw] * (logit[j] * inv);
    fused[idx] = f;
}

// ---------------------------------------------------------------------------
// Launch
// ---------------------------------------------------------------------------
extern "C" void kernel_launch(void* const* d_in, const int* in_sizes, int n_in,
                              void* d_out, int out_size, void* d_ws, size_t ws_size,
                              hipStream_t stream) {
    (void)in_sizes; (void)n_in; (void)out_size; (void)ws_size;
    const float* x     = (const float*)d_in[0];
    const float* la_w  = (const float*)d_in[1];
    const float* la_s  = (const float*)d_in[2];
    const float* la_b  = (const float*)d_in[3];
    const float* ds_w  = (const float*)d_in[4];
    const float* ds_s  = (const float*)d_in[5];
    const float* ds_b  = (const float*)d_in[6];
    const float* lp_w  = (const float*)d_in[7];
    const float* lp_s  = (const float*)d_in[8];
    const float* lp_b  = (const float*)d_in[9];
    const float* hp_w  = (const float*)d_in[10];
    const float* hp_s  = (const float*)d_in[11];
    const float* hp_b  = (const float*)d_in[12];
    const float* fg1_w = (const float*)d_in[13];
    const float* fg1_s = (const float*)d_in[14];
    const float* fg1_b = (const float*)d_in[15];
    const float* fg2_w = (const float*)d_in[16];
    const float* fg2_b = (const float*)d_in[17];
    const float* fr1_w = (const float*)d_in[18];
    const float* fr1_s = (const float*)d_in[19];
    const float* fr1_b = (const float*)d_in[20];
    const float* fr2_w = (const float*)d_in[21];
    const float* fr2_s = (const float*)d_in[22];
    const float* fr2_b = (const float*)d_in[23];
    const float* out_w = (const float*)d_in[24];
    const float* out_s = (const float*)d_in[25];
    const float* out_b = (const float*)d_in[26];
    const float* freq_scale = (const float*)d_in[27];
    const float* res_scale  = (const float*)d_in[28];

    const size_t HW = 65536;   // 256*256
    const size_t hw = 16384;   // 128*128
    float* ws = (float*)d_ws;
    float* pooled   = ws;                     // (4,128,256,256) 33,554,432 f
    float* la_out   = ws + 33554432;          // (4,128,256,256) 33,554,432 f
    float* llhf     = ws + 67108864;          // (4,256,128,128) 16,777,216 f
    float* freq_ctx = ws + 83886080;          // (4,384,128,128) 25,165,824 f
    float* cand     = pooled;                 // reuse: (4,512,128,128)
    float* fg1_out  = ws + 109051904;         // (4,128,128,128)  8,388,608 f
    float* fused    = ws + 117440512;         // (4,128,128,128)  8,388,608 f
    float* fr1_out  = freq_ctx;               // reuse (freq_ctx dead after fg1)
    float* residual = ws + 125829120;         // (4,128,128,128)  8,388,608 f

    // 1) avgpool 3x3
    {
        long total = (long)4 * 128 * HW;
        fls_pool3_kernel<<<(unsigned)((total + 255) / 256), 256, 0, stream>>>(x, pooled, total);
    }
    // 2) la: 1x1 conv 128->128 @ 256x256 + BN + SiLU   (M=128,K=128,N=65536)
    fls_wmma_gemm_cbs<128, false><<<dim3(512, 2, 4), 256, 0, stream>>>(
        la_w, pooled, nullptr, nullptr, la_s, la_b, la_out, 128, (int)HW, 1);
    // 3) Haar: ll/hf + coarse
    {
        long total = (long)4 * 128 * hw;
        fls_haar_kernel<<<(unsigned)((total + 255) / 256), 256, 0, stream>>>(x, llhf, freq_ctx, total);
    }
    // 4) lp: grouped 1x1 on ll -> freq_ctx[0:128]
    fls_wmma_gconv_cbs<16, 1, 1, 0><<<dim3(64, 8, 4), 256, 0, stream>>>(
        lp_w, llhf, 256, 128, 128, lp_s, lp_b, freq_ctx, 384, 0, 128, 128, 1);
    // 5) hp: grouped 1x1 on hf -> freq_ctx[128:256]  (base shifted by 128 ch)
    fls_wmma_gconv_cbs<16, 1, 1, 0><<<dim3(64, 8, 4), 256, 0, stream>>>(
        hp_w, llhf + (size_t)128 * hw, 256, 128, 128, hp_s, hp_b, freq_ctx, 384, 128, 128, 128, 1);
    // 6) ds: grouped 3x3 stride-2 pad-1 conv 128->512 -> cand
    fls_wmma_gconv_cbs<64, 3, 2, 1><<<dim3(128, 8, 4), 256, 0, stream>>>(
        ds_w, x, 128, 256, 256, ds_s, ds_b, cand, 512, 0, 128, 128, 1);
    // 7) fg1: 1x1 conv 384->128 + BN + SiLU   (M=128,K=384,N=16384)
    fls_wmma_gemm_cbs<384, false><<<dim3(128, 2, 4), 256, 0, stream>>>(
        fg1_w, freq_ctx, nullptr, nullptr, fg1_s, fg1_b, fg1_out, 128, (int)hw, 1);
    // 8) attention fuse (fg2 VALU + softmax + cand reduce)
    {
        long total = (long)4 * 128 * hw;
        fls_att_fuse_kernel<<<(unsigned)((total + 255) / 256), 256, 0, stream>>>(
            la_out, fg1_out, cand, fg2_w, fg2_b, freq_scale, fused, total);
    }
    // 9) fr1: 1x1 conv 256->128 + BN + SiLU   (M=128,K=256,N=16384)
    fls_wmma_gemm_cbs<256, false><<<dim3(128, 2, 4), 256, 0, stream>>>(
        fr1_w, llhf, nullptr, nullptr, fr1_s, fr1_b, fr1_out, 128, (int)hw, 1);
    // 10) fr2: grouped 3x3 pad-1 conv 128->128 -> residual
    fls_wmma_gconv_cbs<16, 3, 1, 1><<<dim3(64, 8, 4), 256, 0, stream>>>(
        fr2_w, fr1_out, 128, 128, 128, fr2_s, fr2_b, residual, 128, 0, 128, 128, 1);
    // 11) out: 1x1 conv 128->256 on (fused + res_scale*residual) + BN + SiLU
    fls_wmma_gemm_cbs<128, true><<<dim3(128, 4, 4), 256, 0, stream>>>(
        out_w, fused, residual, res_scale, out_s, out_b, (float*)d_out,
        256, (int)hw, 1);
}